// SIRLayer_5798205850080
// MI455X (gfx1250) — compile-verified
//
#include <hip/hip_runtime.h>

// SIR layer: out[0:N] = beta copy; out[N] = new_I; out[N+1] = new_R
// where r = prob_I @ beta, fused with the streaming copy of beta.
// N = 8192, dt = 0.1, NU = 0.1.

#define SIR_N 8192
#define ROWS_PER_BLOCK 256   // K-rows handled per block (per grid.y)
#define COLS_PER_BLOCK 256   // columns per block (8 waves * 32 cols)
#define ROW_BLOCKS (SIR_N / ROWS_PER_BLOCK)   // 32
#define COL_BLOCKS (SIR_N / COLS_PER_BLOCK)   // 32

typedef __attribute__((ext_vector_type(2))) float v2f;
typedef __attribute__((ext_vector_type(8))) float v8f;

// Main fused kernel: stream beta once; per element do copy-store and WMMA
// accumulate of prob_I[k] * beta[k][col] into per-(rowblock, col) partials.
__global__ __launch_bounds__(256) void sir_main_kernel(
    const float* __restrict__ X,   // (N+2, N)
    float* __restrict__ out,       // (N+2, N)
    float* __restrict__ ws)        // (ROW_BLOCKS, N) partial column sums
{
    const int N = SIR_N;
    const int lane = threadIdx.x & 31;
    const int wave = threadIdx.x >> 5;        // 0..7
    const int half = lane >> 4;               // 0 or 1 (K subgroup)
    const int lcol = lane & 15;               // column within 16-wide tile

    const int cbase = blockIdx.x * COLS_PER_BLOCK + wave * 32;
    const int row0  = blockIdx.y * ROWS_PER_BLOCK;

    const float* __restrict__ beta  = X;
    const float* __restrict__ probI = X + (size_t)N * N;   // row N

    const size_t rstride = (size_t)N;

    // B-layout lane pointers: VGPR0 <- row (k + 2*half), VGPR1 <- row (k + 2*half + 1)
    const float* bptr0 = beta + (size_t)(row0 + 2 * half) * rstride + (size_t)(cbase + lcol);
    const float* bptr1 = bptr0 + 16;  // second 16-column tile
    float* optr0 = out + (size_t)(row0 + 2 * half) * rstride + (size_t)(cbase + lcol);
    float* optr1 = optr0 + 16;
    // A-layout: lanes 0-15 need {p[k], p[k+1]}, lanes 16-31 need {p[k+2], p[k+3]}
    const float* pptr = probI + row0 + 2 * half;

    v8f c0 = {};
    v8f c1 = {};

#pragma unroll 4
    for (int kk = 0; kk < ROWS_PER_BLOCK; kk += 4) {
        // A matrix: prob_I values replicated across all 16 M-rows.
        v2f a;
        a.x = pptr[0];
        a.y = pptr[1];

        // B matrices: two 4x16 f32 tiles (columns cbase..cbase+31).
        v2f b0, b1;
        b0.x = __builtin_nontemporal_load(bptr0);
        b0.y = __builtin_nontemporal_load(bptr0 + rstride);
        b1.x = __builtin_nontemporal_load(bptr1);
        b1.y = __builtin_nontemporal_load(bptr1 + rstride);

        // D = A x B + C : every row of D = partial column sums (A rows replicated)
        c0 = __builtin_amdgcn_wmma_f32_16x16x4_f32(
            false, a, false, b0, (short)0, c0, false, false);
        c1 = __builtin_amdgcn_wmma_f32_16x16x4_f32(
            false, a, false, b1, (short)0, c1, false, false);

        // Fused beta copy: stores mirror the loads exactly (same offsets).
        __builtin_nontemporal_store(b0.x, optr0);
        __builtin_nontemporal_store(b0.y, optr0 + rstride);
        __builtin_nontemporal_store(b1.x, optr1);
        __builtin_nontemporal_store(b1.y, optr1 + rstride);

        bptr0 += 4 * rstride;
        bptr1 += 4 * rstride;
        optr0 += 4 * rstride;
        optr1 += 4 * rstride;
        pptr  += 4;
    }

    // C/D layout: VGPR0, lanes 0-15 hold (M=0, N=lane). All rows identical,
    // so lanes 0-15 of c[0] carry the 16 column sums of each tile.
    if (lane < 16) {
        float* wsrow = ws + (size_t)blockIdx.y * N;
        wsrow[cbase + lcol]      = c0[0];
        wsrow[cbase + 16 + lcol] = c1[0];
    }
}

// Epilogue: reduce ROW_BLOCKS partials per column, apply SIR update.
__global__ __launch_bounds__(256) void sir_epilogue_kernel(
    const float* __restrict__ X,
    const float* __restrict__ ws,
    float* __restrict__ out)
{
    const int N = SIR_N;
    const int j = blockIdx.x * blockDim.x + threadIdx.x;
    if (j >= N) return;

    float r = 0.0f;
#pragma unroll
    for (int rb = 0; rb < ROW_BLOCKS; ++rb) {
        r += ws[(size_t)rb * N + j];
    }

    const float dt = 0.1f;
    const float nu = 0.1f;
    const float pI = X[(size_t)N * N + j];
    const float pR = X[(size_t)(N + 1) * N + j];

    const float susceptible = 1.0f - pI - pR;
    const float trans_rate  = r * susceptible;
    const float trans_prob  = 1.0f - expf(-trans_rate * dt);
    const float recov_prob  = 1.0f - expf(-(nu * pI) * dt);

    out[(size_t)N * N + j]       = pI + trans_prob - recov_prob;
    out[(size_t)(N + 1) * N + j] = pR + recov_prob;
}

extern "C" void kernel_launch(void* const* d_in, const int* in_sizes, int n_in,
                              void* d_out, int out_size, void* d_ws, size_t ws_size,
                              hipStream_t stream) {
    (void)in_sizes; (void)n_in; (void)out_size; (void)ws_size;
    const float* X = (const float*)d_in[0];
    float* out = (float*)d_out;
    float* ws  = (float*)d_ws;   // needs ROW_BLOCKS * N * 4 = 1 MB

    dim3 grid(COL_BLOCKS, ROW_BLOCKS);   // 32 x 32
    sir_main_kernel<<<grid, 256, 0, stream>>>(X, out, ws);
    sir_epilogue_kernel<<<SIR_N / 256, 256, 0, stream>>>(X, ws, out);
}